// FieldTrans_33990371181221
// MI455X (gfx1250) — compile-verified
//
#include <hip/hip_runtime.h>
#include <math.h>

#define Bb 32
#define Ll 256
#define DEPTHd 4
#define Ww 32
#define KK 96          // 3*W stacked K dimension: [center | ortho | diag]
#define HALFX 128      // columns per workgroup in mid-layer kernel

typedef __attribute__((ext_vector_type(2))) float v2f;
typedef __attribute__((ext_vector_type(8))) float v8f;

__device__ __forceinline__ float gelu_exact(float v) {
    return 0.5f * v * (1.0f + erff(v * 0.7071067811865475f));
}

// ---------------- layer 0: 1 -> W (rank-1 weights), sign folds f(+x)/f(-x) ----------------
__global__ __launch_bounds__(256)
void k_first(const float* __restrict__ x, const float* __restrict__ ci,
             const float* __restrict__ bi, const float* __restrict__ ai,
             const float* __restrict__ bias_i, float* __restrict__ h, float sign)
{
    int gid = blockIdx.x * 256 + threadIdx.x;          // b*L*L + y*L + x
    int b  = gid >> 16;
    int y  = (gid >> 8) & 255;
    int xx = gid & 255;
    const float* xb = x + (size_t)b * (Ll * Ll);
    int ym = (y + 255) & 255, yp = (y + 1) & 255;
    int xm = (xx + 255) & 255, xp = (xx + 1) & 255;
    float ctr = xb[y * Ll + xx];
    float nb  = xb[ym * Ll + xx] + xb[yp * Ll + xx] + xb[y * Ll + xm] + xb[y * Ll + xp];
    float dg  = xb[ym * Ll + xm] + xb[ym * Ll + xp] + xb[yp * Ll + xm] + xb[yp * Ll + xp];
    ctr *= sign; nb *= sign; dg *= sign;
    size_t outbase = (size_t)b * Ww * (Ll * Ll) + (size_t)y * Ll + xx;
    #pragma unroll 4
    for (int o = 0; o < Ww; ++o) {
        float v = ci[o] * ctr + bi[o] * nb + ai[o] * dg + bias_i[o];
        h[outbase + (size_t)o * (Ll * Ll)] = gelu_exact(v);
    }
}

// ---------------- mid layers: W -> W, stencil + 32x96 GEMM via f32 WMMA ----------------
// grid: B * L * 2 workgroups; each WG = 256 threads (8 waves), one row, 128 columns.
__global__ __launch_bounds__(256)
void k_mid(const float* __restrict__ hin, const float* __restrict__ wC,
           const float* __restrict__ wB, const float* __restrict__ wA,
           const float* __restrict__ bias, float* __restrict__ hout)
{
    extern __shared__ float smem[];
    float* sS = smem;                 // KK x HALFX stacked stencil matrix (96*128)
    float* sW = smem + KK * HALFX;    // Ww x KK concatenated weights (32*96)

    const int t  = threadIdx.x;
    const int r  = blockIdx.x;                // ((b*256 + y)*2 + hx)
    const int b  = r >> 9;
    const int y  = (r >> 1) & 255;
    const int hx = r & 1;
    const int xbase = hx * HALFX;

    // stage Wcat[o][k]: k = part*32 + ch; part 0=center(wC), 1=ortho(wB), 2=diag(wA)
    for (int idx = t; idx < Ww * KK; idx += 256) {
        int o = idx / KK, k = idx - o * KK;
        int part = k >> 5, ch = k & 31;
        float v = (part == 0) ? wC[o * Ww + ch]
                : (part == 1) ? wB[o * Ww + ch]
                              : wA[o * Ww + ch];
        sW[idx] = v;
    }

    // build S[k][xloc] for this row segment (periodic 9-point stencil)
    const float* hb = hin + (size_t)b * Ww * (Ll * Ll);
    const int ym = (y + 255) & 255, yp = (y + 1) & 255;
    const int xloc = t & 127;
    const int chbase = (t >> 7) * 16;          // threads 0..127 -> ch 0..15, 128..255 -> 16..31
    const int xg = xbase + xloc;
    const int xm = (xg + 255) & 255, xp = (xg + 1) & 255;
    #pragma unroll 4
    for (int cc = 0; cc < 16; ++cc) {
        int ch = chbase + cc;
        const float* hc = hb + (size_t)ch * (Ll * Ll);
        float c = hc[y * Ll + xg];
        float n = hc[ym * Ll + xg] + hc[yp * Ll + xg] + hc[y * Ll + xm] + hc[y * Ll + xp];
        float d = hc[ym * Ll + xm] + hc[ym * Ll + xp] + hc[yp * Ll + xm] + hc[yp * Ll + xp];
        sS[(0 * Ww + ch) * HALFX + xloc] = c;
        sS[(1 * Ww + ch) * HALFX + xloc] = n;
        sS[(2 * Ww + ch) * HALFX + xloc] = d;
    }
    __syncthreads();

    // WMMA: D(32 x 16) = Wcat(32x96) * S(96x16) per wave's N-tile
    const int wave = t >> 5, lane = t & 31;
    const int half = lane >> 4, col = lane & 15;
    const int n0 = wave * 16;                  // 8 waves cover 128 columns

    v8f acc0 = {};                             // M rows 0..15
    v8f acc1 = {};                             // M rows 16..31
    for (int kc = 0; kc < KK; kc += 4) {
        const int kA = kc + 2 * half;
        v2f a0, a1, bf;
        // A 16x4: lanes0-15 hold K=kc/kc+1, lanes16-31 hold K=kc+2/kc+3
        a0.x = sW[col * KK + kA];          a0.y = sW[col * KK + kA + 1];
        a1.x = sW[(col + 16) * KK + kA];   a1.y = sW[(col + 16) * KK + kA + 1];
        // B 4x16: same K arrangement across lane halves, N striped over cols
        bf.x = sS[kA * HALFX + n0 + col];  bf.y = sS[(kA + 1) * HALFX + n0 + col];
        acc0 = __builtin_amdgcn_wmma_f32_16x16x4_f32(false, a0, false, bf,
                                                     (short)0, acc0, false, false);
        acc1 = __builtin_amdgcn_wmma_f32_16x16x4_f32(false, a1, false, bf,
                                                     (short)0, acc1, false, false);
    }

    // epilogue: bias + exact GELU, coalesced stores (lanes cover contiguous x)
    float* outb = hout + (size_t)b * Ww * (Ll * Ll) + (size_t)y * Ll;
    const int xo = xbase + n0 + col;
    #pragma unroll
    for (int j = 0; j < 8; ++j) {
        int o0 = j + 8 * half;                 // C/D layout: VGPR j -> M=j (lanes0-15), j+8 (lanes16-31)
        float v0 = acc0[j] + bias[o0];
        float v1 = acc1[j] + bias[o0 + 16];
        outb[(size_t)o0 * (Ll * Ll) + xo]        = gelu_exact(v0);
        outb[(size_t)(o0 + 16) * (Ll * Ll) + xo] = gelu_exact(v1);
    }
}

// ---------------- final layer part 1: contract channels into 3 planes ----------------
__global__ __launch_bounds__(256)
void k_contract(const float* __restrict__ h, const float* __restrict__ co,
                const float* __restrict__ bo, const float* __restrict__ ao,
                float* __restrict__ gC, float* __restrict__ gB, float* __restrict__ gA)
{
    int gid  = blockIdx.x * 256 + threadIdx.x;
    int b    = gid >> 16;
    int rest = gid & 65535;
    const float* hb = h + (size_t)b * Ww * (Ll * Ll) + rest;
    float sc = 0.f, sb = 0.f, sa = 0.f;
    #pragma unroll 4
    for (int i = 0; i < Ww; ++i) {
        float v = hb[(size_t)i * (Ll * Ll)];
        sc += co[i] * v; sb += bo[i] * v; sa += ao[i] * v;
    }
    gC[gid] = sc; gB[gid] = sb; gA[gid] = sa;
}

// ---------------- final layer part 2: stencil combine + antisymmetric accumulate ----------------
__global__ __launch_bounds__(256)
void k_combine(const float* __restrict__ gC, const float* __restrict__ gB,
               const float* __restrict__ gA, float* __restrict__ out, int first)
{
    int gid = blockIdx.x * 256 + threadIdx.x;
    int b  = gid >> 16;
    int y  = (gid >> 8) & 255;
    int xx = gid & 255;
    size_t pb = (size_t)b * (Ll * Ll);
    int ym = (y + 255) & 255, yp = (y + 1) & 255;
    int xm = (xx + 255) & 255, xp = (xx + 1) & 255;
    float v = gC[pb + y * Ll + xx]
            + gB[pb + ym * Ll + xx] + gB[pb + yp * Ll + xx]
            + gB[pb + y * Ll + xm]  + gB[pb + y * Ll + xp]
            + gA[pb + ym * Ll + xm] + gA[pb + ym * Ll + xp]
            + gA[pb + yp * Ll + xm] + gA[pb + yp * Ll + xp];
    if (first) out[gid] = 0.5f * v;            // pass f(+x): initialize
    else       out[gid] -= 0.5f * v;           // pass f(-x): subtract
}

extern "C" void kernel_launch(void* const* d_in, const int* in_sizes, int n_in,
                              void* d_out, int out_size, void* d_ws, size_t ws_size,
                              hipStream_t stream) {
    (void)in_sizes; (void)n_in; (void)out_size; (void)ws_size;
    const float* x      = (const float*)d_in[0];
    const float* ai     = (const float*)d_in[1];   // (W,1) diag, first layer
    const float* ao     = (const float*)d_in[2];   // (1,W) diag, last layer
    const float* aw     = (const float*)d_in[3];   // (DEPTH,W,W) diag
    const float* bi     = (const float*)d_in[4];   // (W,1) ortho, first layer
    const float* bo     = (const float*)d_in[5];   // (1,W) ortho, last layer
    const float* bw     = (const float*)d_in[6];   // (DEPTH,W,W) ortho
    const float* ci     = (const float*)d_in[7];   // (W,1) center, first layer
    const float* co     = (const float*)d_in[8];   // (1,W) center, last layer
    const float* cw     = (const float*)d_in[9];   // (DEPTH,W,W) center
    const float* bias_i = (const float*)d_in[10];
    const float* bias   = (const float*)d_in[11];

    const size_t plane = (size_t)Ll * Ll;               // 65536
    const size_t hsz   = (size_t)Bb * Ww * plane;       // 67.1M floats (256 MB)
    float* hA = (float*)d_ws;
    float* hB = hA + hsz;
    float* gC = hB + hsz;
    float* gB = gC + (size_t)Bb * plane;
    float* gA = gB + (size_t)Bb * plane;

    const int gridElem = (Bb * Ll * Ll) / 256;          // 8192
    const int gridMid  = Bb * Ll * 2;                   // 16384
    const size_t ldsMid = (size_t)(KK * HALFX + Ww * KK) * sizeof(float);  // 61440 B
    dim3 blk(256);

    for (int s = 0; s < 2; ++s) {
        float sign = (s == 0) ? 1.f : -1.f;
        k_first<<<gridElem, blk, 0, stream>>>(x, ci, bi, ai, bias_i, hA, sign);
        const float* cur = hA; float* nxt = hB;
        for (int k = 0; k < DEPTHd; ++k) {
            k_mid<<<gridMid, blk, ldsMid, stream>>>(cur,
                    cw + k * Ww * Ww, bw + k * Ww * Ww, aw + k * Ww * Ww,
                    bias + k * Ww, nxt);
            float* tmp = (float*)cur; cur = nxt; nxt = tmp;
        }
        k_contract<<<gridElem, blk, 0, stream>>>(cur, co, bo, ao, gC, gB, gA);
        k_combine<<<gridElem, blk, 0, stream>>>(gC, gB, gA, (float*)d_out, s == 0 ? 1 : 0);
    }
}